// LossLD_49684181680790
// MI455X (gfx1250) — compile-verified
//
#include <hip/hip_runtime.h>
#include <math.h>

// ---------------------------------------------------------------------------
// KL( softmax(gaussian_code(label)) || softmax(scores) ), batchmean.
// Per row i:  S(mu_i) + logsumexp(scores_i) - dot(T[mu_i], scores_i)
// where T is the 100x100 label->target table and S(mu)=sum_j T log T.
// Memory-bound: scores read exactly once, coalesced 128b loads.
// ---------------------------------------------------------------------------

#define NCLASS   100
#define THREADS  256
#define WAVES    (THREADS / 32)
#define NBLOCKS  1024              // multiple of 64 for the WMMA final reduce

typedef float v2f __attribute__((ext_vector_type(2)));
typedef float v8f __attribute__((ext_vector_type(8)));

__global__ void __launch_bounds__(THREADS)
loss_main(const float* __restrict__ scores,
          const int*   __restrict__ labels,
          float*       __restrict__ partials,
          int n)
{
    __shared__ float sT[NCLASS * NCLASS];  // target table, row-major
    __shared__ float sS[NCLASS];           // per-mu  sum t*log t
    __shared__ float sWave[WAVES];

    const int tid = threadIdx.x;

    // ---- build target table: thread mu (< 100) computes one row ----------
    if (tid < NCLASS) {
        const float inv_norm = 0.39894228040143267794f;  // 1/sqrt(2*pi)
        const float mu = (float)tid;
        float* trow = &sT[tid * NCLASS];
        // pass 1: store exp(code_j), accumulate sum (code in [0,0.399]: safe)
        float sum = 0.0f;
        #pragma unroll 4
        for (int j = 0; j < NCLASS; ++j) {
            float d = (float)j - mu;
            float code = inv_norm * expf(-0.5f * d * d);
            float e = expf(code);
            trow[j] = e;
            sum += e;
        }
        const float rsum   = 1.0f / sum;
        const float logsum = logf(sum);
        // pass 2: normalize, accumulate entropy term  sum t*(code - logsum)
        float S = 0.0f;
        #pragma unroll 4
        for (int j = 0; j < NCLASS; ++j) {
            float e = trow[j];
            float t = e * rsum;
            S += t * (logf(e) - logsum);   // log t = code - logsum
            trow[j] = t;
        }
        sS[tid] = S;
    }
    __syncthreads();

    // ---- stream rows: one wave32 per row ----------------------------------
    const int wave   = tid >> 5;
    const int lane   = tid & 31;
    const int gwave  = blockIdx.x * WAVES + wave;
    const int nwaves = gridDim.x * WAVES;
    const bool active = (lane < 25);       // 25 lanes * 4 floats = 100 cols

    float acc = 0.0f;
    for (int row = gwave; row < n; row += nwaves) {
        int lbl = labels[row];
        lbl = (lbl < 0) ? 0 : ((lbl >= NCLASS) ? NCLASS - 1 : lbl);

        const float* srow = scores + (size_t)row * NCLASS;

        float4 s4 = make_float4(0.f, 0.f, 0.f, 0.f);
        float  m  = -3.0e38f;
        if (active) {
            s4 = *(const float4*)(srow + lane * 4);   // 400B row, 16B aligned
            m  = fmaxf(fmaxf(s4.x, s4.y), fmaxf(s4.z, s4.w));
        }
        #pragma unroll
        for (int off = 16; off >= 1; off >>= 1)
            m = fmaxf(m, __shfl_xor(m, off, 32));

        float se = 0.0f, dt = 0.0f;
        if (active) {
            se = expf(s4.x - m) + expf(s4.y - m) +
                 expf(s4.z - m) + expf(s4.w - m);
            const float* trow = &sT[lbl * NCLASS + lane * 4];
            dt = fmaf(s4.x, trow[0],
                 fmaf(s4.y, trow[1],
                 fmaf(s4.z, trow[2], s4.w * trow[3])));
        }
        #pragma unroll
        for (int off = 16; off >= 1; off >>= 1) {
            se += __shfl_xor(se, off, 32);
            dt += __shfl_xor(dt, off, 32);
        }

        acc += sS[lbl] + (m + logf(se)) - dt;   // identical in all lanes
    }

    if (lane == 0) sWave[wave] = acc;
    __syncthreads();
    if (tid == 0) {
        float b = 0.0f;
        #pragma unroll
        for (int w = 0; w < WAVES; ++w) b += sWave[w];
        partials[blockIdx.x] = b;
    }
}

// ---------------------------------------------------------------------------
// Deterministic final reduction of NBLOCKS partials via V_WMMA_F32_16X16X4_F32.
// C += A(16x4 of partials) x B(4x16 of ones).  D[m,n] = rowsum_m, so the
// grand total = sum over column 0 = (lane0: c[0..7]) + (lane16: c[0..7]).
// Single wave32, EXEC all ones, no divergence around the WMMA.
// ---------------------------------------------------------------------------
__global__ void __launch_bounds__(32)
loss_final(const float* __restrict__ partials, float* __restrict__ out,
           float inv_n)
{
    const int lane = threadIdx.x;

    v8f c = {};                 // accumulator = 0
    v2f b; b.x = 1.0f; b.y = 1.0f;   // all-ones B (layout-invariant)

    // A layout (32-bit 16x4, ISA 7.12.2): lanes 0-15 -> {K=0,K=1},
    // lanes 16-31 -> {K=2,K=3}; map A[m,k] = partials[base + 4m + k].
    const int sub = (lane < 16) ? (4 * lane) : (4 * (lane - 16) + 2);
    #pragma unroll
    for (int base = 0; base < NBLOCKS; base += 64) {
        v2f a;
        a.x = partials[base + sub];
        a.y = partials[base + sub + 1];
        c = __builtin_amdgcn_wmma_f32_16x16x4_f32(
                /*neg_a=*/false, a, /*neg_b=*/false, b,
                /*c_mod=*/(short)0, c, /*reuse_a=*/false, /*reuse_b=*/false);
    }

    // column 0 of D: lane 0 holds D[0..7,0], lane 16 holds D[8..15,0]
    float s = c[0] + c[1] + c[2] + c[3] + c[4] + c[5] + c[6] + c[7];
    float total = __shfl(s, 0, 32) + __shfl(s, 16, 32);
    if (lane == 0) out[0] = total * inv_n;
}

extern "C" void kernel_launch(void* const* d_in, const int* in_sizes, int n_in,
                              void* d_out, int out_size, void* d_ws, size_t ws_size,
                              hipStream_t stream)
{
    const float* scores  = (const float*)d_in[0];
    const int*   labels  = (const int*)d_in[1];
    float*       out     = (float*)d_out;
    float*       partials = (float*)d_ws;          // NBLOCKS floats = 4 KB

    const int n = in_sizes[1];                      // labels count = rows

    loss_main<<<NBLOCKS, THREADS, 0, stream>>>(scores, labels, partials, n);
    loss_final<<<1, 32, 0, stream>>>(partials, out, 1.0f / (float)n);
}